// TorchNeighborListRangeSeparated_19258633355895
// MI455X (gfx1250) — compile-verified
//
#include <hip/hip_runtime.h>
#include <hip/hip_bf16.h>

typedef __attribute__((ext_vector_type(2))) float v2f;
typedef __attribute__((ext_vector_type(8))) float v8f;

// ---------------------------------------------------------------------------
// Kernel 1: shifted-image block. One wave32 per 16x16 (i,j) tile per shift.
//   d2(i,j) = |r_i|^2 + |q_j|^2 - 2 r_i . q_j,   q_j = r_j - shiftvec
// Cross term + |r_i|^2 folded into one V_WMMA_F32_16X16X4_F32:
//   A row i = (x_i, y_i, z_i, |r_i|^2),  B col j = (-2qx, -2qy, -2qz, 1)
// Grid: x = j-tile groups (8 waves/block), y = i-tile, z = shift. No div/mod.
// NC = compile-time number of cutoffs -> fully unrolled mask stores.
// Interior tiles (wave-uniform test) take a branch-free strength-reduced
// store path; only boundary tiles pay per-element guards.
// ---------------------------------------------------------------------------
template <int NC>
__global__ void __launch_bounds__(256)
nbr_tiles_wmma(const float* __restrict__ pos,
               const float* __restrict__ cell,
               const float* __restrict__ shifts,
               const float* __restrict__ cutoffs,
               float* __restrict__ out,
               int n, int TJ, long long Pc, long long P)
{
    const int tj = blockIdx.x * (blockDim.x >> 5) + (threadIdx.x >> 5);
    if (tj >= TJ) return;                         // wave-uniform: EXEC stays full

    const int ti = blockIdx.y;
    const int s  = blockIdx.z;

    const int lane = threadIdx.x & 31;
    const int col  = lane & 15;
    const bool hi  = lane >= 16;

    // shiftvec = shifts[s] @ cell  (row-vector times matrix); scalar-cached loads
    const float s0 = shifts[s * 3 + 0], s1 = shifts[s * 3 + 1], s2 = shifts[s * 3 + 2];
    const float sv0 = s0 * cell[0] + s1 * cell[3] + s2 * cell[6];
    const float sv1 = s0 * cell[1] + s1 * cell[4] + s2 * cell[7];
    const float sv2 = s0 * cell[2] + s1 * cell[5] + s2 * cell[8];

    // ---- A (16x4 f32): lanes 0-15 -> (K0,K1)=(x,y); lanes 16-31 -> (K2,K3)=(z,|r|^2)
    const int i0 = ti * 16 + col;
    const int ic = (i0 < n) ? i0 : (n - 1);       // clamp loads; stores are guarded
    const float ax = pos[ic * 3 + 0];
    const float ay = pos[ic * 3 + 1];
    const float az = pos[ic * 3 + 2];
    const float an2 = ax * ax + ay * ay + az * az;
    v2f A;
    A.x = hi ? az  : ax;
    A.y = hi ? an2 : ay;

    // ---- B (4x16 f32): lanes 0-15 -> rows K0,K1; lanes 16-31 -> rows K2,K3
    const int j0 = tj * 16 + col;
    const int jc = (j0 < n) ? j0 : (n - 1);
    const float qx = pos[jc * 3 + 0] - sv0;
    const float qy = pos[jc * 3 + 1] - sv1;
    const float qz = pos[jc * 3 + 2] - sv2;
    const float qn2 = qx * qx + qy * qy + qz * qz;
    v2f B;
    B.x = hi ? (-2.0f * qz) : (-2.0f * qx);
    B.y = hi ? 1.0f         : (-2.0f * qy);

    // ---- D = A x B + 0  -> v_wmma_f32_16x16x4_f32
    v8f C = {};
    C = __builtin_amdgcn_wmma_f32_16x16x4_f32(
            /*neg_a=*/false, A, /*neg_b=*/false, B,
            /*c_mod=*/(short)0, C, /*reuse_a=*/false, /*reuse_b=*/false);

    float cut2[NC];
#pragma unroll
    for (int c = 0; c < NC; ++c) { const float cv = cutoffs[c]; cut2[c] = cv * cv; }

    // C layout: VGPR v -> M = v + (hi ? 8 : 0), N = lane & 15
    const long long rowBase = Pc + (long long)s * n * n;   // + i*n + j
    const int irow0 = ti * 16 + (hi ? 8 : 0);

    // Wave-uniform interior test: whole 16x16 tile in bounds?
    if (ti * 16 + 16 <= n && tj * 16 + 16 <= n) {
        // Fast path: no guards, incremental addressing, clause-friendly stores.
        float* ptr[NC + 1];
        ptr[0] = out + (rowBase + (long long)irow0 * n + j0);
#pragma unroll
        for (int c = 0; c < NC; ++c) ptr[c + 1] = ptr[0] + (long long)(c + 1) * P;
#pragma unroll
        for (int v = 0; v < 8; ++v) {
            const float d2 = C[v] + qn2;                   // add |q_j|^2
            ptr[0][0] = d2;
#pragma unroll
            for (int c = 0; c < NC; ++c)
                ptr[c + 1][0] = (d2 < cut2[c]) ? 1.0f : 0.0f;
#pragma unroll
            for (int k = 0; k <= NC; ++k) ptr[k] += n;     // next row
        }
    } else {
        // Boundary tiles: per-element guards.
        const bool jok = (j0 < n);
#pragma unroll
        for (int v = 0; v < 8; ++v) {
            const int irow = irow0 + v;
            if (jok && irow < n) {
                const long long p = rowBase + (long long)irow * n + j0;
                const float d2 = C[v] + qn2;
                out[p] = d2;
#pragma unroll
                for (int c = 0; c < NC; ++c)
                    out[P + (long long)c * P + p] = (d2 < cut2[c]) ? 1.0f : 0.0f;
            }
        }
    }
}

// ---------------------------------------------------------------------------
// Kernel 2: in-cell upper-triangular pairs (i<j), one thread per pair.
// Pair index p -> (i, j): offset(i) = i*n - i*(i+1)/2
// ---------------------------------------------------------------------------
__device__ __forceinline__ long long triu_off(long long i, long long n) {
    return i * n - (i * (i + 1)) / 2;
}

template <int NC>
__global__ void __launch_bounds__(256)
nbr_triu(const float* __restrict__ pos,
         const float* __restrict__ cutoffs,
         float* __restrict__ out,
         int n, long long Pc, long long P)
{
    const long long p = (long long)blockIdx.x * blockDim.x + threadIdx.x;
    if (p >= Pc) return;

    const double nn = (double)n;
    const double disc = (2.0 * nn - 1.0) * (2.0 * nn - 1.0) - 8.0 * (double)p;
    long long i = (long long)((2.0 * nn - 1.0 - sqrt(disc)) * 0.5);
    if (i < 0) i = 0;
    if (i > n - 2) i = n - 2;
    while (i > 0 && triu_off(i, n) > p) --i;
    while (i < n - 2 && triu_off(i + 1, n) <= p) ++i;
    const long long j = p - triu_off(i, n) + i + 1;

    const float dx = pos[i * 3 + 0] - pos[j * 3 + 0];
    const float dy = pos[i * 3 + 1] - pos[j * 3 + 1];
    const float dz = pos[i * 3 + 2] - pos[j * 3 + 2];
    const float d2 = dx * dx + dy * dy + dz * dz;

    out[p] = d2;
#pragma unroll
    for (int c = 0; c < NC; ++c) {
        const float cv = cutoffs[c];
        out[P + (long long)c * P + p] = (d2 < cv * cv) ? 1.0f : 0.0f;
    }
}

// Generic fallback for unexpected cutoff counts (no compile-time unroll).
__global__ void __launch_bounds__(256)
nbr_masks_generic(const float* __restrict__ cutoffs, float* __restrict__ out,
                  int ncut, long long P)
{
    const long long p = (long long)blockIdx.x * blockDim.x + threadIdx.x;
    if (p >= P) return;
    const float d2 = out[p];
    for (int c = 0; c < ncut; ++c) {
        const float cv = cutoffs[c];
        out[P + (long long)c * P + p] = (d2 < cv * cv) ? 1.0f : 0.0f;
    }
}

// ---------------------------------------------------------------------------
template <int NC>
static void launch_all(const float* pos, const float* cell, const float* shifts,
                       const float* cutoffs, float* out,
                       int n, int S, long long Pc, long long P, hipStream_t stream)
{
    const int TI = (n + 15) / 16;
    const int TJ = (n + 15) / 16;
    const int wavesPerBlock = 256 / 32;
    dim3 grid((TJ + wavesPerBlock - 1) / wavesPerBlock, TI, S);
    nbr_tiles_wmma<NC><<<grid, 256, 0, stream>>>(
        pos, cell, shifts, cutoffs, out, n, TJ, Pc, P);

    const int blocksTriu = (int)((Pc + 255) / 256);
    nbr_triu<NC><<<blocksTriu, 256, 0, stream>>>(pos, cutoffs, out, n, Pc, P);
}

extern "C" void kernel_launch(void* const* d_in, const int* in_sizes, int n_in,
                              void* d_out, int out_size, void* d_ws, size_t ws_size,
                              hipStream_t stream)
{
    const float* pos     = (const float*)d_in[0];   // [n,3]
    const float* cell    = (const float*)d_in[1];   // [3,3]
    const float* shifts  = (const float*)d_in[2];   // [S,3]
    const float* cutoffs = (const float*)d_in[3];   // [ncut]
    float* out = (float*)d_out;

    const int n    = in_sizes[0] / 3;
    const int S    = in_sizes[2] / 3;
    const int ncut = in_sizes[3];

    const long long Pc = (long long)n * (n - 1) / 2;
    const long long P  = Pc + (long long)S * n * n;

    switch (ncut) {
    case 1: launch_all<1>(pos, cell, shifts, cutoffs, out, n, S, Pc, P, stream); break;
    case 2: launch_all<2>(pos, cell, shifts, cutoffs, out, n, S, Pc, P, stream); break;
    case 3: launch_all<3>(pos, cell, shifts, cutoffs, out, n, S, Pc, P, stream); break;
    case 4: launch_all<4>(pos, cell, shifts, cutoffs, out, n, S, Pc, P, stream); break;
    default: {
        // write d2 only (NC=1-style), then generic mask pass over all P pairs
        launch_all<1>(pos, cell, shifts, cutoffs, out, n, S, Pc, P, stream);
        const int blocks = (int)((P + 255) / 256);
        nbr_masks_generic<<<blocks, 256, 0, stream>>>(cutoffs, out, ncut, P);
        break;
    }
    }
}